// SSMLayer_33174327395017
// MI455X (gfx1250) — compile-verified
//
#include <hip/hip_runtime.h>

// SSM layer (S4-style) for gfx1250 / MI455X.
// Chunked parallel scan: all heavy math is f32 WMMA GEMMs
// (V_WMMA_F32_16X16X4_F32); sequential depth is O(log L) GEMM launches.
//
// Sizes: L = 2^20, N = 64, chunk S = 1024, chunks Q = 1024.

#define L_SEQ (1 << 20)
#define NLAT 64
#define SCH 1024
#define QCH 1024

typedef __attribute__((ext_vector_type(2))) float v2f;
typedef __attribute__((ext_vector_type(4))) float v4f;
typedef __attribute__((ext_vector_type(8))) float v8f;
typedef int ssm_v4i __attribute__((vector_size(16)));  // matches builtin param

#define AS1 __attribute__((address_space(1)))
#define AS3 __attribute__((address_space(3)))

#if defined(__has_builtin)
#if __has_builtin(__builtin_amdgcn_global_load_async_to_lds_b128) && \
    __has_builtin(__builtin_amdgcn_s_wait_asynccnt)
#define SSM_HAVE_ASYNC_LDS 1
#endif
#endif

// ---------------------------------------------------------------------------
// Kernel 1: bilinear discretization.  One block, 256 threads.
//   G = I - (dt/2) A  ;  Bl = G^-1 (Gauss-Jordan; G is strongly diag-dominant)
//   Abar = Bl (I + (dt/2) A)  -> Apow slot 0
//   Bbar = Bl (dt B)
// ---------------------------------------------------------------------------
__global__ __launch_bounds__(256)
void ssm_discretize_kernel(const float* __restrict__ A,
                           const float* __restrict__ Bv,
                           const float* __restrict__ dtp,
                           float* __restrict__ Apow0,
                           float* __restrict__ Bbar) {
    __shared__ float G[NLAT * 129];   // [64 x 128] augmented, padded stride 129
    __shared__ float colp[NLAT];
    const int tid = threadIdx.x;
    const float dt = dtp[0];
    const float h = 0.5f * dt;

    for (int i = tid; i < NLAT * NLAT; i += 256) {
        int r = i >> 6, c = i & 63;
        G[r * 129 + c]      = ((r == c) ? 1.0f : 0.0f) - h * A[i];
        G[r * 129 + 64 + c] = (r == c) ? 1.0f : 0.0f;
    }
    __syncthreads();

    for (int p = 0; p < NLAT; ++p) {
        if (tid < NLAT) colp[tid] = G[tid * 129 + p];
        __syncthreads();
        float inv = 1.0f / colp[p];
        if (tid < 128) G[p * 129 + tid] *= inv;
        __syncthreads();
        for (int i = tid; i < NLAT * 128; i += 256) {
            int r = i >> 7, c = i & 127;
            if (r != p) G[r * 129 + c] -= colp[r] * G[p * 129 + c];
        }
        __syncthreads();
    }

    // Abar = Bl + h * (Bl @ A)
    for (int i = tid; i < NLAT * NLAT; i += 256) {
        int r = i >> 6, c = i & 63;
        float s = 0.0f;
        for (int t = 0; t < NLAT; ++t)
            s += G[r * 129 + 64 + t] * A[t * NLAT + c];
        Apow0[i] = G[r * 129 + 64 + c] + h * s;
    }
    if (tid < NLAT) {
        float s = 0.0f;
        for (int t = 0; t < NLAT; ++t)
            s += G[tid * 129 + 64 + t] * Bv[t];
        Bbar[tid] = dt * s;
    }
}

// ---------------------------------------------------------------------------
// Kernel 2: Apow[k] = Abar^(2^k), k = 1..19 (repeated squaring in LDS).
// ---------------------------------------------------------------------------
__global__ __launch_bounds__(256)
void ssm_powers_kernel(float* __restrict__ Apow) {
    __shared__ float bufA[NLAT * NLAT];
    __shared__ float bufB[NLAT * NLAT];
    const int tid = threadIdx.x;
    for (int i = tid; i < NLAT * NLAT; i += 256) bufA[i] = Apow[i];
    __syncthreads();
    float* src = bufA;
    float* dst = bufB;
    for (int k = 1; k < 20; ++k) {
        for (int i = tid; i < NLAT * NLAT; i += 256) {
            int r = i >> 6, c = i & 63;
            float acc = 0.0f;
            for (int t = 0; t < NLAT; ++t)
                acc += src[r * NLAT + t] * src[t * NLAT + c];
            dst[i] = acc;
        }
        __syncthreads();
        for (int i = tid; i < NLAT * NLAT; i += 256) Apow[k * NLAT * NLAT + i] = dst[i];
        float* tmp = src; src = dst; dst = tmp;
        __syncthreads();
    }
}

// ---------------------------------------------------------------------------
// Kernel 3: log-doubling of V (cols Abar^m Bbar) and Wd (rows C Abar^{r+1}).
// Block 0 -> V (64 x S row-major), block 1 -> Wd (S x 64 row-major).
// ---------------------------------------------------------------------------
__global__ __launch_bounds__(256)
void ssm_doubling_kernel(const float* __restrict__ Apow,
                         const float* __restrict__ Bbar,
                         const float* __restrict__ Cv,
                         float* __restrict__ V,
                         float* __restrict__ Wd) {
    __shared__ float Ak[NLAT * NLAT];
    const int tid = threadIdx.x;
    if (blockIdx.x == 0) {
        if (tid < NLAT) V[tid * SCH] = Bbar[tid];   // V[:,0] = Bbar
        __threadfence();
        __syncthreads();
        for (int k = 0; k < 10; ++k) {
            const int m0 = 1 << k;
            for (int i = tid; i < NLAT * NLAT; i += 256) Ak[i] = Apow[k * NLAT * NLAT + i];
            __syncthreads();
            for (int i = tid; i < NLAT * m0; i += 256) {
                int n = i >> k;
                int m = i & (m0 - 1);
                float acc = 0.0f;
                for (int t = 0; t < NLAT; ++t)
                    acc += Ak[n * NLAT + t] * V[t * SCH + m];
                V[n * SCH + m0 + m] = acc;
            }
            __threadfence();
            __syncthreads();
        }
    } else {
        if (tid < NLAT) {
            float acc = 0.0f;
            for (int t = 0; t < NLAT; ++t)
                acc += Cv[t] * Apow[t * NLAT + tid];
            Wd[tid] = acc;    // Wd[0,:] = C @ Abar
        }
        __threadfence();
        __syncthreads();
        for (int k = 0; k < 10; ++k) {
            const int m0 = 1 << k;
            for (int i = tid; i < NLAT * NLAT; i += 256) Ak[i] = Apow[k * NLAT * NLAT + i];
            __syncthreads();
            for (int i = tid; i < m0 * NLAT; i += 256) {
                int r = i >> 6;
                int n = i & 63;
                float acc = 0.0f;
                for (int t = 0; t < NLAT; ++t)
                    acc += Wd[r * NLAT + t] * Ak[t * NLAT + n];
                Wd[(m0 + r) * NLAT + n] = acc;
            }
            __threadfence();
            __syncthreads();
        }
    }
}

// ---------------------------------------------------------------------------
// Kernel 4: K[m] = C . V[:,m]
// ---------------------------------------------------------------------------
__global__ __launch_bounds__(256)
void ssm_kv_kernel(const float* __restrict__ Cv,
                   const float* __restrict__ V,
                   float* __restrict__ Kv) {
    const int m = blockIdx.x * 256 + threadIdx.x;
    float acc = 0.0f;
    for (int n = 0; n < NLAT; ++n)
        acc += Cv[n] * V[n * SCH + m];
    Kv[m] = acc;
}

// ---------------------------------------------------------------------------
// Kernel 5: T[r][j] = K[r-j] for j <= r else 0
// ---------------------------------------------------------------------------
__global__ __launch_bounds__(256)
void ssm_tfill_kernel(const float* __restrict__ Kv,
                      float* __restrict__ T) {
    const int idx = blockIdx.x * 256 + threadIdx.x;
    const int r = idx >> 10;
    const int j = idx & (SCH - 1);
    T[idx] = (j <= r) ? Kv[r - j] : 0.0f;
}

// ---------------------------------------------------------------------------
// Kernel 6: tiled transpose: U[j][q] = u[q*S + j]
// ---------------------------------------------------------------------------
__global__ void ssm_transpose_kernel(const float* __restrict__ u,
                                     float* __restrict__ U) {
    __shared__ float tile[32][33];
    const int qBase = blockIdx.x * 32;
    const int jBase = blockIdx.y * 32;
    for (int i = threadIdx.y; i < 32; i += 8)
        tile[i][threadIdx.x] = u[(qBase + i) * SCH + (jBase + threadIdx.x)];
    __syncthreads();
    for (int i = threadIdx.y; i < 32; i += 8)
        U[(jBase + i) * QCH + (qBase + threadIdx.x)] = tile[threadIdx.x][i];
}

// ---------------------------------------------------------------------------
// Kernel 7: generic f32 WMMA GEMM for the SMALL GEMMs.
//   D = (Cin or 0) + A @ B'    A: MxK (lda), B: KxN (ldb), row-major.
//   shift: B column index is (n - shift); columns < 0 contribute 0.
//   revb:  B row index is (K-1-k).
// One wave per 16x16 output tile.  B loads are branch-free: clamped address +
// cndmask-zero (no exec save/restore in the inner loop).
// ---------------------------------------------------------------------------
__global__ __launch_bounds__(256)
void ssm_wmma_gemm_f32(const float* __restrict__ A, int lda,
                       const float* __restrict__ B, int ldb,
                       const float* __restrict__ Cin, int ldc,
                       float* __restrict__ D, int ldd,
                       int M, int N, int K, int shift, int revb) {
    const int w = blockIdx.x * (blockDim.x >> 5) + (threadIdx.x >> 5);
    const int tilesN = N >> 4;
    const int tilesM = M >> 4;
    if (w >= tilesM * tilesN) return;
    const int tm = w / tilesN;
    const int tn = w % tilesN;
    const int lane = threadIdx.x & 31;
    const int half = lane >> 4;
    const int l16  = lane & 15;
    const int arow = tm * 16 + l16;
    const int bcol = tn * 16 + l16;
    const int c2   = bcol - shift;
    const int cc   = (c2 < 0) ? 0 : c2;      // clamped, always in-bounds
    const float bm = (c2 < 0) ? 0.0f : 1.0f; // mask via v_cndmask, no branch

    v8f acc = {};
    if (Cin) {
#pragma unroll
        for (int i = 0; i < 8; ++i)
            acc[i] = Cin[(tm * 16 + half * 8 + i) * ldc + bcol];
    }

    const float* Arow = A + arow * lda;
#pragma unroll 4
    for (int k = 0; k < K; k += 4) {
        const int kk = k + half * 2;
        v2f a;
        a.x = Arow[kk];
        a.y = Arow[kk + 1];
        const int kr0 = revb ? (K - 1 - kk) : kk;
        const int kr1 = revb ? (K - 2 - kk) : (kk + 1);
        v2f b;
        b.x = B[kr0 * ldb + cc] * bm;
        b.y = B[kr1 * ldb + cc] * bm;
        acc = __builtin_amdgcn_wmma_f32_16x16x4_f32(
            false, a, false, b, (short)0, acc, false, false);
    }

#pragma unroll
    for (int i = 0; i < 8; ++i)
        D[(tm * 16 + half * 8 + i) * ldd + bcol] = acc[i];
}

// ---------------------------------------------------------------------------
// Kernel 8: LDS-tiled WMMA GEMM for the heavy step:  D += A @ B
//   Fixed 1024x1024x1024, row-major.  Block = 64x64 macro-tile, 8 waves.
//   Wave (wm in 0..3, wn in 0..1) computes two 16x16 tiles sharing its
//   A fragment.  A/B chunks (KB=32) staged in LDS; padded strides 36/68
//   floats keep 16B row alignment (for b128 copies) and spread LDS banks.
//   Staging uses global_load_async_to_lds_b128 when the builtin exists.
// ---------------------------------------------------------------------------
#define KB 32
#define ASTRIDE 36   // floats; 144B rows (16B aligned), bank-friendly
#define BSTRIDE 68   // floats; 272B rows (16B aligned)

__global__ __launch_bounds__(256)
void ssm_wmma_gemm_tiled(const float* __restrict__ A,
                         const float* __restrict__ B,
                         float* __restrict__ D) {
    __shared__ float Ast[64 * ASTRIDE];  // Ast[r*ASTRIDE + k] = A[bm*64+r][k0+k]
    __shared__ float Bs [KB * BSTRIDE];  // Bs [k*BSTRIDE + n] = B[k0+k][bn*64+n]

    const int bn = blockIdx.x;
    const int bm = blockIdx.y;
    const int tid  = threadIdx.x;
    const int wid  = tid >> 5;
    const int wm   = wid & 3;        // 4 wave rows
    const int wn   = wid >> 2;       // 2 wave cols (x2 tiles each)
    const int lane = tid & 31;
    const int half = lane >> 4;
    const int l16  = lane & 15;
    const int arow = wm * 16 + l16;
    const int bc0  = wn * 32 + l16;

    const float* Ag = A + (size_t)(bm * 64) * 1024;
    const float* Bg = B + bn * 64;

    // Accumulators initialized from D (acts as the C matrix: D += A@B).
    const int r0  = bm * 64 + wm * 16 + half * 8;
    const int cN0 = bn * 64 + wn * 32 + l16;
    v8f acc0 = {}, acc1 = {};
#pragma unroll
    for (int i = 0; i < 8; ++i) {
        acc0[i] = D[(size_t)(r0 + i) * 1024 + cN0];
        acc1[i] = D[(size_t)(r0 + i) * 1024 + cN0 + 16];
    }

    for (int k0 = 0; k0 < 1024; k0 += KB) {
        // ---- stage A (64 x KB) and B (KB x 64) chunks into LDS ----
#ifdef SSM_HAVE_ASYNC_LDS
#pragma unroll
        for (int s = tid; s < 512; s += 256) {   // 512 float4 slots of A
            const int r = s >> 3, c4 = s & 7;
            __builtin_amdgcn_global_load_async_to_lds_b128(
                (AS1 ssm_v4i*)(Ag + (size_t)r * 1024 + k0 + c4 * 4),
                (AS3 ssm_v4i*)(&Ast[r * ASTRIDE + c4 * 4]), 0, 0);
        }
#pragma unroll
        for (int s = tid; s < 512; s += 256) {   // 512 float4 slots of B
            const int r = s >> 4, c4 = s & 15;
            __builtin_amdgcn_global_load_async_to_lds_b128(
                (AS1 ssm_v4i*)(Bg + (size_t)(k0 + r) * 1024 + c4 * 4),
                (AS3 ssm_v4i*)(&Bs[r * BSTRIDE + c4 * 4]), 0, 0);
        }
        __builtin_amdgcn_s_wait_asynccnt(0);
#else
#pragma unroll
        for (int s = tid; s < 512; s += 256) {
            const int r = s >> 3, c4 = s & 7;
            v4f v = *(const v4f*)(Ag + (size_t)r * 1024 + k0 + c4 * 4);
            *(v4f*)(&Ast[r * ASTRIDE + c4 * 4]) = v;
        }
#pragma unroll
        for (int s = tid; s < 512; s += 256) {
            const int r = s >> 4, c4 = s & 15;
            v4f v = *(const v4f*)(Bg + (size_t)(k0 + r) * 1024 + c4 * 4);
            *(v4f*)(&Bs[r * BSTRIDE + c4 * 4]) = v;
        }
#endif
        __syncthreads();

        // ---- 8 k-quads x 2 WMMA, A fragment reused across both N tiles ----
#pragma unroll
        for (int kq = 0; kq < KB; kq += 4) {
            const int kk = kq + half * 2;
            v2f a;
            a.x = Ast[arow * ASTRIDE + kk];
            a.y = Ast[arow * ASTRIDE + kk + 1];
            v2f b0, b1;
            b0.x = Bs[kk * BSTRIDE + bc0];
            b0.y = Bs[(kk + 1) * BSTRIDE + bc0];
            b1.x = Bs[kk * BSTRIDE + bc0 + 16];
            b1.y = Bs[(kk + 1) * BSTRIDE + bc0 + 16];
            acc0 = __builtin_amdgcn_wmma_f32_16x16x4_f32(
                false, a, false, b0, (short)0, acc0, false, false);
            acc1 = __builtin_amdgcn_wmma_f32_16x16x4_f32(
                false, a, false, b1, (short)0, acc1, false, false);
        }
        __syncthreads();
    }

#pragma unroll
    for (int i = 0; i < 8; ++i) {
        D[(size_t)(r0 + i) * 1024 + cN0]      = acc0[i];
        D[(size_t)(r0 + i) * 1024 + cN0 + 16] = acc1[i];
    }
}

// ---------------------------------------------------------------------------
// Kernel 9: transpose-combine: y[q*S + r] = Yacc[r][q] + D0 * u[q*S + r]
// ---------------------------------------------------------------------------
__global__ void ssm_combine_kernel(const float* __restrict__ Yacc,
                                   const float* __restrict__ u,
                                   const float* __restrict__ Dp,
                                   float* __restrict__ y) {
    __shared__ float tile[32][33];
    const int rBase = blockIdx.x * 32;
    const int qBase = blockIdx.y * 32;
    for (int i = threadIdx.y; i < 32; i += 8)
        tile[i][threadIdx.x] = Yacc[(rBase + i) * QCH + (qBase + threadIdx.x)];
    __syncthreads();
    const float D0 = Dp[0];
    for (int i = threadIdx.y; i < 32; i += 8) {
        const int t = (qBase + i) * SCH + (rBase + threadIdx.x);
        y[t] = tile[threadIdx.x][i] + D0 * u[t];
    }
}

// ---------------------------------------------------------------------------
static inline void launch_gemm(const float* A, int lda, const float* B, int ldb,
                               const float* Cin, int ldc, float* D, int ldd,
                               int M, int N, int K, int shift, int revb,
                               hipStream_t s) {
    const int tiles = (M / 16) * (N / 16);
    const int blocks = (tiles + 7) / 8;   // 8 waves (256 threads) per block
    ssm_wmma_gemm_f32<<<blocks, 256, 0, s>>>(A, lda, B, ldb, Cin, ldc, D, ldd,
                                             M, N, K, shift, revb);
}

extern "C" void kernel_launch(void* const* d_in, const int* in_sizes, int n_in,
                              void* d_out, int out_size, void* d_ws, size_t ws_size,
                              hipStream_t stream) {
    const float* u   = (const float*)d_in[0];   // (L,)
    const float* Ain = (const float*)d_in[1];   // (64,64)
    const float* Bin = (const float*)d_in[2];   // (64,1)
    const float* Cin = (const float*)d_in[3];   // (1,64)
    const float* Dp  = (const float*)d_in[4];   // scalar
    const float* dtp = (const float*)d_in[5];   // scalar
    float* y = (float*)d_out;                   // (L,)

    float* ws = (float*)d_ws;
    // Workspace layout (floats); total ~3.49M floats (~14 MB).
    float* Apow = ws;                       // 20 * 4096
    float* Bbar = Apow + 20 * 4096;         // 64
    float* V    = Bbar + 64;                // 64 * 1024
    float* Wd   = V + NLAT * SCH;           // 1024 * 64
    float* Kv   = Wd + SCH * NLAT;          // 1024
    float* T    = Kv + SCH;                 // 1024 * 1024
    float* U    = T + SCH * SCH;            // 1024 * 1024
    float* c0   = U + SCH * QCH;            // 64 * 1024
    float* c1   = c0 + NLAT * QCH;          // 64 * 1024
    float* Yacc = c1 + NLAT * QCH;          // 1024 * 1024

    // 1. Discretize: Abar -> Apow[0], Bbar.
    ssm_discretize_kernel<<<1, 256, 0, stream>>>(Ain, Bin, dtp, Apow, Bbar);
    // 2. Apow[k] = Abar^(2^k), k = 1..19.
    ssm_powers_kernel<<<1, 256, 0, stream>>>(Apow);
    // 3. V (Abar^m Bbar columns) and Wd (C Abar^{r+1} rows) by doubling.
    ssm_doubling_kernel<<<2, 256, 0, stream>>>(Apow, Bbar, Cin, V, Wd);
    // 4. K[m] = C . V[:,m]
    ssm_kv_kernel<<<SCH / 256, 256, 0, stream>>>(Cin, V, Kv);
    // 5. T lower-triangular Toeplitz from K.
    ssm_tfill_kernel<<<(SCH * SCH) / 256, 256, 0, stream>>>(Kv, T);
    // 6. U[j][q] = u[q*S + j]
    ssm_transpose_kernel<<<dim3(QCH / 32, SCH / 32), dim3(32, 8), 0, stream>>>(u, U);

    // 7. Per-chunk aggregates: c0 = b = V @ revrows(U)   (64 x Q, K = S)
    launch_gemm(V, SCH, U, QCH, nullptr, 0, c0, QCH, NLAT, QCH, SCH,
                /*shift=*/0, /*revb=*/1, stream);

    // 8. Hillis-Steele scan over chunks: c <- c + Abar^(S*2^k) @ shift(c, 2^k)
    for (int k = 0; k < 10; ++k) {
        const float* src = (k & 1) ? c1 : c0;
        float*       dst = (k & 1) ? c0 : c1;
        launch_gemm(Apow + (10 + k) * 4096, NLAT, src, QCH, src, QCH, dst, QCH,
                    NLAT, QCH, NLAT, /*shift=*/(1 << k), /*revb=*/0, stream);
    }
    // cfinal now in c0; chunk-start state of chunk q is cfinal[:, q-1]
    // (0 for q == 0) -> handled via shift=1 in Y1.

    // 9. Y1 = Wd @ states    (S x Q, K = 64)
    launch_gemm(Wd, NLAT, c0, QCH, nullptr, 0, Yacc, QCH, SCH, QCH, NLAT,
                /*shift=*/1, /*revb=*/0, stream);
    // 10. Yacc += T @ U      (S x Q, K = S)  -- heavy GEMM, LDS-tiled WMMA
    ssm_wmma_gemm_tiled<<<dim3(16, 16), 256, 0, stream>>>(T, U, Yacc);

    // 11. y[t] = Yacc[r][q] + D*u[t]
    ssm_combine_kernel<<<dim3(SCH / 32, QCH / 32), dim3(32, 8), 0, stream>>>(
        Yacc, u, Dp, y);

    (void)in_sizes; (void)n_in; (void)out_size; (void)ws_size;
}